// iidBindingSbm_68839735820335
// MI455X (gfx1250) — compile-verified
//
#include <hip/hip_runtime.h>
#include <math.h>

#define NB 64
#define NUM_EPS 1e-8f

typedef __attribute__((ext_vector_type(2))) float v2f;
typedef __attribute__((ext_vector_type(8))) float v8f;

// ---------------------------------------------------------------------------
// Kernel 1: block-level precompute (a, R, P_rem), E_m, isolated-node count.
// One block, 64 threads; thread b owns block-row b.
// Exploits SBM block structure: the 4096x4096 per-row argsort in
// _isolated_nodes collapses exactly to a 64-element block-level sort because
// tied (ep, sp) groups telescope order-independently.
// ---------------------------------------------------------------------------
__global__ __launch_bounds__(NB) void sbm_precompute(
    const float* __restrict__ alpha,
    const float* __restrict__ p_blocks,
    const int*   __restrict__ N_blocks,
    const int*   __restrict__ n_round_p,
    float* __restrict__ wa, float* __restrict__ wN,
    float* __restrict__ wR, float* __restrict__ wP,
    float* __restrict__ out)
{
    __shared__ float a_sh[NB];
    __shared__ float n_sh[NB];
    __shared__ float red[NB];
    const int b = threadIdx.x;
    const float nr = (float)(*n_round_p);
    const float inv_nr = 1.0f / nr;

    const float ab = 1.0f / (1.0f + expf(-alpha[b]));   // sigmoid
    const float Nb = (float)N_blocks[b];
    a_sh[b] = ab; n_sh[b] = Nb;
    wa[b] = ab;   wN[b] = Nb;
    __syncthreads();

    float rrow[NB];
    float em_part  = 0.0f;   // sum_c p[b,c]*Nb*Nc  (minus diag elements)
    float prod_rem = 1.0f;   // prod_c (1 - P_rem[b,c])^{m_c}
    for (int c = 0; c < NB; ++c) {
        const float p  = p_blocks[b * NB + c];
        const float ac = a_sh[c];
        const float Q  = ab * ac;
        float R  = (1.0f - powf(1.0f - p, inv_nr)) / (Q + NUM_EPS);
        R = fminf(fmaxf(R, 0.0f), 1.0f);
        float Pr = 1.0f - (1.0f - p) / (powf(1.0f - Q, nr) + NUM_EPS);
        Pr = fminf(fmaxf(Pr, 0.0f), 1.0f);
        wR[b * NB + c] = R;                  // block-level R (diag NOT zeroed; matches triplet path)
        wP[b * NB + c] = Pr;
        rrow[c] = R;
        const float m = (c == b) ? (Nb - 1.0f) : n_sh[c];   // diag element excluded
        prod_rem *= powf(1.0f - Pr, m);
        em_part += p * Nb * n_sh[c];
        if (c == b) em_part -= p * Nb;       // remove diagonal entries of P_full
    }

    // Descending selection-sort over 64 block values (exact collapse of the
    // 4096-element per-row sorted cumprod in _isolated_nodes).
    float acc = 0.0f, cp = 1.0f;
    for (int t = 0; t < NB; ++t) {
        int best = 0; float bv = -1.0f;
        for (int c = 0; c < NB; ++c) {
            const float v = rrow[c];
            if (v > bv) { bv = v; best = c; }
        }
        rrow[best] = -2.0f;
        const float m = (best == b) ? (Nb - 1.0f) : n_sh[best];
        const float g = powf(1.0f - a_sh[best], m);   // group survival prod(1-sp)
        acc += cp * bv * (1.0f - g);
        cp  *= g;
    }
    const float pc  = ab * acc;
    const float pi  = powf(1.0f - pc, nr);
    const float iso = Nb * pi * prod_rem;

    // Three deterministic tree reductions over 64 threads.
    red[b] = em_part; __syncthreads();
    for (int s = 32; s > 0; s >>= 1) { if (b < s) red[b] += red[b + s]; __syncthreads(); }
    const float em_total = red[0];
    __syncthreads();
    red[b] = iso; __syncthreads();
    for (int s = 32; s > 0; s >>= 1) { if (b < s) red[b] += red[b + s]; __syncthreads(); }
    const float iso_total = red[0];
    __syncthreads();
    red[b] = Nb; __syncthreads();
    for (int s = 32; s > 0; s >>= 1) { if (b < s) red[b] += red[b + s]; __syncthreads(); }
    const float n_total = red[0];

    if (b == 0) {
        out[4] = 0.5f * em_total;        // E_m
        out[5] = n_total - iso_total;    // n_nonisolated
    }
}

// ---------------------------------------------------------------------------
// Kernel 2: the 64^3 triplet grid. Block (i,j) = blockIdx, thread = k.
// Deterministic shared-memory tree reduction per block -> 4 partials.
// ---------------------------------------------------------------------------
#define CSWAP3(x0,y0,z0,x1,y1,z1) { float t_; t_=x0;x0=x1;x1=t_; t_=y0;y0=y1;y1=t_; t_=z0;z0=z1;z1=t_; }

__global__ __launch_bounds__(NB) void sbm_triplets(
    const float* __restrict__ wa, const float* __restrict__ wN,
    const float* __restrict__ wR, const float* __restrict__ wP,
    const int*   __restrict__ n_round_p,
    float* __restrict__ part)
{
    __shared__ float sh0[NB], sh1[NB], sh2[NB], sh3[NB];
    const int bid = blockIdx.x;
    const int i = bid >> 6;
    const int j = bid & 63;
    const int k = threadIdx.x;
    const float nr = (float)(*n_round_p);

    // Stack order from reference: [ (R_ij,Pr_ij,a_k), (R_ik,Pr_ik,a_j), (R_jk,Pr_jk,a_i) ]
    float rA = wR[i*NB + j], rB = wR[i*NB + k], rC = wR[j*NB + k];
    float pA = wP[i*NB + j], pB = wP[i*NB + k], pC = wP[j*NB + k];
    float sA = wa[k],        sB = wa[j],        sC = wa[i];

    // Stable ascending sort by r (matches jnp.argsort stability on ties).
    if (rB < rA) CSWAP3(rA,pA,sA, rB,pB,sB);
    if (rC < rB) { CSWAP3(rB,pB,sB, rC,pC,sC); if (rB < rA) CSWAP3(rA,pA,sA, rB,pB,sB); }

    const float Rmin = rA, Rmid = rB, Rmax = rC;
    const float Prmin = pA, Prmid = pB, Prmax = pC;
    const float amin = sA, amid = sB, amax = sC;

    const float pg3    = amax * amid * amin;
    const float pg2max = (1.0f - amax) * amid * amin;
    const float pg2mid = amax * (1.0f - amid) * amin;
    const float pg2min = amax * amid * (1.0f - amin);

    const float pr3    = pg3 * Rmin;
    const float pr2    = pg3 * (Rmid - Rmin);
    const float pr1max = pg3 * (Rmax - Rmid) + pg2max * Rmax;
    const float pr1mid = pg2mid * Rmid;
    const float pr1min = pg2min * Rmin;
    const float pr0    = 1.0f - pr3 - pr2 - pr1max - pr1mid - pr1min;

    const float q0       = powf(pr0, nr);
    const float q1max    = powf(pr0 + pr1max, nr);
    const float q1mid    = powf(pr0 + pr1mid, nr);
    const float q1min    = powf(pr0 + pr1min, nr);
    const float q2maxmid = powf(pr0 + pr1max + pr1mid + pr2, nr);
    const float q2maxmin = powf(pr0 + pr1max + pr1min, nr);
    const float q2midmin = powf(pr0 + pr1mid + pr1min, nr);

    const float omx = 1.0f - Prmax, omd = 1.0f - Prmid, omn = 1.0f - Prmin;

    const float P0    = q0 * omx * omd * omn;
    const float P1max = q0 * Prmax * omd * omn + (q1max - q0) * omd * omn;
    const float P1mid = q0 * Prmid * omn * omx + (q1mid - q0) * omn * omx;
    const float P1min = q0 * Prmin * omx * omd + (q1min - q0) * omx * omd;
    const float P2mm  = q0 * Prmax * Prmid * omn + (q1max - q0) * Prmid * omn
                      + (q1mid - q0) * Prmax * omn + (q2maxmid - q1max - q1mid + q0) * omn;
    const float P2mn  = q0 * Prmax * Prmin * omd + (q1max - q0) * Prmin * omd
                      + (q1min - q0) * Prmax * omd + (q2maxmin - q1max - q1min + q0) * omd;
    const float P2dn  = q0 * Prmid * Prmin * omx + (q1mid - q0) * Prmin * omx
                      + (q1min - q0) * Prmid * omx + (q2midmin - q1mid - q1min + q0) * omx;
    const float P3    = 1.0f - P0 - P1max - P1mid - P1min - P2mm - P2mn - P2dn;
    const float P1    = P1max + P1mid + P1min;
    const float P2    = P2mm + P2mn + P2dn;

    const float Ni = wN[i], Nj = wN[j], Nk = wN[k];
    float w;
    if (i == j && j == k)   w = Ni * (Ni - 1.0f) * (Ni - 2.0f) * (1.0f / 6.0f);
    else if (i == j)        w = Ni * (Ni - 1.0f) * Nk * (1.0f / 6.0f);
    else if (i == k)        w = Ni * (Ni - 1.0f) * Nj * (1.0f / 6.0f);
    else if (j == k)        w = Nj * (Nj - 1.0f) * Ni * (1.0f / 6.0f);
    else                    w = Ni * Nj * Nk * (1.0f / 6.0f);

    sh0[k] = P0 * w; sh1[k] = P1 * w; sh2[k] = P2 * w; sh3[k] = P3 * w;
    __syncthreads();
    for (int s = 32; s > 0; s >>= 1) {
        if (k < s) { sh0[k] += sh0[k+s]; sh1[k] += sh1[k+s]; sh2[k] += sh2[k+s]; sh3[k] += sh3[k+s]; }
        __syncthreads();
    }
    if (k == 0) {
        float* o = part + bid * 4;
        o[0] = sh0[0]; o[1] = sh1[0]; o[2] = sh2[0]; o[3] = sh3[0];
    }
}

// ---------------------------------------------------------------------------
// Kernel 3: single-wave final reduction of 4096 partial vectors (4 floats).
// Cross-lane sum done with V_WMMA_F32_16X16X4_F32: D = A(16x4) x ones(4x16).
// A layout (ISA 7.12.2, 32-bit A 16x4): VGPR0 = K0 (lanes 0-15) / K2 (16-31),
// VGPR1 = K1/K3. Setting A[0]=acc, A[1]=0 in every lane gives
// rowsum_m = acc(lane m) + acc(lane m+16) replicated across all D columns.
// D layout: lane<16 VGPR r = D[r, lane]; lane>=16 VGPR r = D[r+8, lane-16].
// So sum of a lane's 8 D regs + shfl_xor(16) = full 32-lane sum. One wave of
// exactly 32 threads, straight-line code => EXEC all-1s as WMMA requires.
// Fixed summation order every call => deterministic.
// ---------------------------------------------------------------------------
__global__ __launch_bounds__(32) void sbm_reduce_wmma(
    const float* __restrict__ part, float* __restrict__ out, int nparts)
{
    const int lane = threadIdx.x;
    float acc0 = 0.f, acc1 = 0.f, acc2 = 0.f, acc3 = 0.f;
    for (int idx = lane; idx < nparts; idx += 32) {   // nparts % 32 == 0: no divergence
        const float* p = part + idx * 4;
        acc0 += p[0]; acc1 += p[1]; acc2 += p[2]; acc3 += p[3];
    }
    const v2f Bones = {1.0f, 1.0f};   // B = all-ones 4x16 (layout-independent)
    float accs[4] = {acc0, acc1, acc2, acc3};
    float tot[4];
#pragma unroll
    for (int e = 0; e < 4; ++e) {
        v2f A = {accs[e], 0.0f};
        v8f C = {0.f, 0.f, 0.f, 0.f, 0.f, 0.f, 0.f, 0.f};
        v8f D = __builtin_amdgcn_wmma_f32_16x16x4_f32(
            /*neg_a=*/false, A, /*neg_b=*/false, Bones,
            /*c_mod=*/(short)0, C, /*reuse_a=*/false, /*reuse_b=*/false);
        float s = D[0] + D[1] + D[2] + D[3] + D[4] + D[5] + D[6] + D[7];
        s += __shfl_xor(s, 16, 32);
        tot[e] = s;
    }
    if (lane == 0) { out[0] = tot[0]; out[1] = tot[1]; out[2] = tot[2]; out[3] = tot[3]; }
}

// ---------------------------------------------------------------------------
extern "C" void kernel_launch(void* const* d_in, const int* in_sizes, int n_in,
                              void* d_out, int out_size, void* d_ws, size_t ws_size,
                              hipStream_t stream)
{
    const float* alpha    = (const float*)d_in[0];
    const float* p_blocks = (const float*)d_in[1];
    const int*   N_blocks = (const int*)d_in[2];
    const int*   n_round  = (const int*)d_in[3];
    float* out = (float*)d_out;

    // Workspace layout (floats): a[64] | N[64] | R[64*64] | P_rem[64*64] | partials[4096*4]
    float* ws   = (float*)d_ws;
    float* wa   = ws;
    float* wN   = ws + NB;
    float* wR   = ws + 2 * NB;
    float* wP   = wR + NB * NB;
    float* partials = wP + NB * NB;   // 16384 floats; total ~97 KB of d_ws

    sbm_precompute<<<1, NB, 0, stream>>>(alpha, p_blocks, N_blocks, n_round,
                                         wa, wN, wR, wP, out);
    sbm_triplets<<<NB * NB, NB, 0, stream>>>(wa, wN, wR, wP, n_round, partials);
    sbm_reduce_wmma<<<1, 32, 0, stream>>>(partials, out, NB * NB);
}